// TextRenderingLoss_39333310497234
// MI455X (gfx1250) — compile-verified
//
#include <hip/hip_runtime.h>
#include <math.h>

typedef float v2f __attribute__((ext_vector_type(2)));
typedef float v8f __attribute__((ext_vector_type(8)));

#define B_  32
#define C_  4
#define H_  512
#define W_  512
#define T_  64
#define BOX_ 5
#define ROWCHUNKS 16
#define ROWS_PER_CHUNK (H_ / ROWCHUNKS)            // 32
#define NBLOCKS (B_ * C_ * ROWCHUNKS)              // 2048
#define NTOTAL  ((double)B_ * C_ * H_ * W_)        // 33554432

// ---------------------------------------------------------------------------
// Kernel 1: build 64-bit token coverage masks per row / per column, per batch.
// rowTok[b*H + h] bit t set  <=>  token t valid && h in [ypos-5, ypos+5)
// colTok[b*W + w] bit t set  <=>  token t valid && w in [xpos-5, xpos+5)
// mask[b,h,w] = (rowTok & colTok) != 0  — exactly the reference's box union.
// ---------------------------------------------------------------------------
__global__ __launch_bounds__(512) void trl_mask_kernel(
    const float* __restrict__ pos,          // [B, T, 2] (x, y)
    unsigned long long* __restrict__ rowTok,
    unsigned long long* __restrict__ colTok) {
  __shared__ float sx[T_];
  __shared__ float sy[T_];
  const int b   = blockIdx.x;
  const int tid = threadIdx.x;
  if (tid < T_) {
    sx[tid] = pos[(b * T_ + tid) * 2 + 0];
    sy[tid] = pos[(b * T_ + tid) * 2 + 1];
  }
  __syncthreads();

  unsigned long long r = 0ull, c = 0ull;
  const int i = tid;  // serves as both row index h and col index w (H == W == 512)
#pragma unroll 4
  for (int t = 0; t < T_; ++t) {
    const float x = sx[t], y = sy[t];
    const bool valid = (x > 0.0f) && (y > 0.0f);
    const int ypos = (int)floorf(y * (float)H_);
    const int xpos = (int)floorf(x * (float)W_);
    if (valid && i >= ypos - BOX_ && i < ypos + BOX_) r |= (1ull << t);
    if (valid && i >= xpos - BOX_ && i < xpos + BOX_) c |= (1ull << t);
  }
  rowTok[b * H_ + i] = r;
  colTok[b * W_ + i] = c;
}

// ---------------------------------------------------------------------------
// Kernel 2: streaming weighted SSE. Each block handles a 32-row slab of one
// (b,c) image plane: 32*512 floats x 2 arrays, float4 (b128) coalesced loads.
// Wave-level reduction done with V_WMMA_F32_16X16X4_F32 (B = ones).
// ---------------------------------------------------------------------------
__global__ __launch_bounds__(256) void trl_loss_kernel(
    const float* __restrict__ pred,
    const float* __restrict__ targ,
    const unsigned long long* __restrict__ rowTok,
    const unsigned long long* __restrict__ colTok,
    double* __restrict__ blockSums) {
  __shared__ unsigned long long sCol[W_];             // 4 KB
  __shared__ unsigned long long sRow[ROWS_PER_CHUNK]; // 256 B
  __shared__ float sWave[8];

  const int bid     = blockIdx.x;
  const int chunk   = bid & (ROWCHUNKS - 1);
  const int img     = bid >> 4;          // b*C + c
  const int b       = img >> 2;
  const int baseRow = chunk * ROWS_PER_CHUNK;
  const int tid     = threadIdx.x;

  for (int i = tid; i < W_; i += 256) sCol[i] = colTok[b * W_ + i];
  if (tid < ROWS_PER_CHUNK) sRow[tid] = rowTok[b * H_ + baseRow + tid];
  __syncthreads();

  const size_t base = ((size_t)img * H_ + (size_t)baseRow) * W_;
  const float4* __restrict__ p4 = (const float4*)(pred + base);
  const float4* __restrict__ t4 = (const float4*)(targ + base);

  float acc = 0.0f;
  const int n4 = ROWS_PER_CHUNK * (W_ / 4);  // 4096 float4s per block
  for (int i = tid; i < n4; i += 256) {
    const float4 p = p4[i];
    const float4 t = t4[i];
    const int rl = i >> 7;       // row within slab
    const int wb = (i & 127) << 2;
    const unsigned long long rt = sRow[rl];
    const float w0 = (sCol[wb + 0] & rt) ? 4.0f : 1.0f;  // 1 + 3*mask
    const float w1 = (sCol[wb + 1] & rt) ? 4.0f : 1.0f;
    const float w2 = (sCol[wb + 2] & rt) ? 4.0f : 1.0f;
    const float w3 = (sCol[wb + 3] & rt) ? 4.0f : 1.0f;
    const float d0 = p.x - t.x;
    const float d1 = p.y - t.y;
    const float d2 = p.z - t.z;
    const float d3 = p.w - t.w;
    acc = fmaf(w0 * d0, d0, acc);
    acc = fmaf(w1 * d1, d1, acc);
    acc = fmaf(w2 * d2, d2, acc);
    acc = fmaf(w3 * d3, d3, acc);
  }

  // ---- wave32 reduction via WMMA: D = A(16x4) * ones(4x16) ----
  // A holds the 32 lane partials (2 slots/lane; second slot zero).
  // D[m][n] = rowsum[m] for every n; lanes 0-15 see rows 0-7 in d[0..7],
  // lanes 16-31 see rows 8-15, so s + shfl_xor(s,16) = full wave sum.
  v2f a;  a.x = acc;  a.y = 0.0f;
  v2f bones; bones.x = 1.0f; bones.y = 1.0f;
  v8f cz = {};
  v8f d = __builtin_amdgcn_wmma_f32_16x16x4_f32(
      /*neg_a=*/false, a, /*neg_b=*/false, bones,
      /*c_mod=*/(short)0, cz, /*reuse_a=*/false, /*reuse_b=*/false);
  float s = d[0] + d[1] + d[2] + d[3] + d[4] + d[5] + d[6] + d[7];
  s += __shfl_xor(s, 16, 32);

  const int lane = tid & 31;
  const int wave = tid >> 5;
  if (lane == 0) sWave[wave] = s;
  __syncthreads();
  if (tid == 0) {
    float total = 0.0f;
#pragma unroll
    for (int wv = 0; wv < 8; ++wv) total += sWave[wv];
    blockSums[bid] = (double)total;
  }
}

// ---------------------------------------------------------------------------
// Kernel 3: deterministic final reduction of block sums (no atomics).
// ---------------------------------------------------------------------------
__global__ __launch_bounds__(256) void trl_finalize_kernel(
    const double* __restrict__ blockSums, float* __restrict__ out) {
  __shared__ double s[256];
  const int tid = threadIdx.x;
  double v = 0.0;
  for (int i = tid; i < NBLOCKS; i += 256) v += blockSums[i];
  s[tid] = v;
  __syncthreads();
  for (int stride = 128; stride > 0; stride >>= 1) {
    if (tid < stride) s[tid] += s[tid + stride];
    __syncthreads();
  }
  if (tid == 0) out[0] = (float)(s[0] / NTOTAL);
}

// ---------------------------------------------------------------------------
extern "C" void kernel_launch(void* const* d_in, const int* in_sizes, int n_in,
                              void* d_out, int out_size, void* d_ws, size_t ws_size,
                              hipStream_t stream) {
  const float* pred = (const float*)d_in[0];   // [B,C,H,W] f32
  const float* targ = (const float*)d_in[1];   // [B,C,H,W] f32
  // d_in[2] = text_tokens (int64) — unused by the math
  const float* pos  = (const float*)d_in[3];   // [B,T,2] f32

  // workspace layout: [blockSums: 2048 doubles][rowTok: B*H u64][colTok: B*W u64]
  double* blockSums = (double*)d_ws;
  unsigned long long* rowTok =
      (unsigned long long*)((char*)d_ws + (size_t)NBLOCKS * sizeof(double));
  unsigned long long* colTok = rowTok + (size_t)B_ * H_;

  trl_mask_kernel<<<B_, 512, 0, stream>>>(pos, rowTok, colTok);
  trl_loss_kernel<<<NBLOCKS, 256, 0, stream>>>(pred, targ, rowTok, colTok, blockSums);
  trl_finalize_kernel<<<1, 256, 0, stream>>>(blockSums, (float*)d_out);
}